// SpikeFP32ScaleBy2K_43860206027310
// MI455X (gfx1250) — compile-verified
//
#include <hip/hip_runtime.h>
#include <cstdint>

// ---------------------------------------------------------------------------
// SpikeFP32ScaleBy2K: soft-logic fp32 exponent adjust (x * 2^k on 0/1 "bit"
// floats). Memory-bound: ~768 MiB @ 23.3 TB/s => ~34 us floor. Strategy:
// CDNA5 async global<->LDS DMA (ASYNCcnt) for perfectly coalesced streaming,
// integer bit-pack (inputs are exactly 0x0 / 0x3F800000, so bit 29 == value),
// ~15 integer ops per row, async store back from LDS.
// ---------------------------------------------------------------------------

typedef int v4i __attribute__((vector_size(16)));
typedef __attribute__((address_space(3))) v4i as3_v4i;

__device__ __forceinline__ void async_copy_g2l_b128(const void* gsrc, unsigned lds_off) {
#if __has_builtin(__builtin_amdgcn_global_load_async_to_lds_b128)
    __builtin_amdgcn_global_load_async_to_lds_b128(
        (v4i*)(uintptr_t)gsrc, (as3_v4i*)lds_off, 0, 0);
#else
    asm volatile("global_load_async_to_lds_b128 %0, %1, off"
                 :: "v"(lds_off), "v"((unsigned long long)(uintptr_t)gsrc)
                 : "memory");
#endif
}

__device__ __forceinline__ void async_copy_l2g_b128(void* gdst, unsigned lds_off) {
#if __has_builtin(__builtin_amdgcn_global_store_async_from_lds_b128)
    __builtin_amdgcn_global_store_async_from_lds_b128(
        (v4i*)(uintptr_t)gdst, (as3_v4i*)lds_off, 0, 0);
#else
    asm volatile("global_store_async_from_lds_b128 %0, %1, off"
                 :: "v"((unsigned long long)(uintptr_t)gdst), "v"(lds_off)
                 : "memory");
#endif
}

__device__ __forceinline__ void wait_asynccnt0() {
#if __has_builtin(__builtin_amdgcn_s_wait_asynccnt)
    __builtin_amdgcn_s_wait_asynccnt(0);
#else
    asm volatile("s_wait_asynccnt 0x0" ::: "memory");
#endif
}

// Exact integer equivalent of the reference gate network (bit j of the row is
// bit (31-j) of the packed word => IEEE fp32 layout).
__device__ __forceinline__ unsigned scale_bits(unsigned wx, unsigned wk) {
    unsigned ex   = (wx >> 23) & 0xFFu;               // exponent of x
    unsigned ek   = (wk >> 23) & 0xFFu;               // exponent of k
    unsigned sk   = wk >> 31;                         // sign of k
    unsigned val  = 0x80u | ((wk >> 16) & 0x7Fu);     // hidden bit + top7 mantissa
    unsigned sh3  = (ek + 1u) & 7u;                   // (ek - 127) & 7
    unsigned kabs = val >> (7u - sh3);                // |k| as 8-bit int
    unsigned kfin = sk ? ((0u - kabs) & 0xFFu) : kabs;
    unsigned enew = (ex + kfin) & 0xFFu;
    unsigned scl  = (wx & 0x807FFFFFu) | (enew << 23);
    return ((wk & 0x7FFFFFFFu) == 0u) ? wx : scl;     // k==0 -> passthrough
}

#define ONE_F32 0x3F800000u

#define TROWS 128          // rows per block == threads per block
#define PADB  144          // padded LDS row stride (36 dwords -> conflict-free b128)
#define TILEB (TROWS * PADB)

__global__ __launch_bounds__(TROWS)
void spike_scale_tile_kernel(const float* __restrict__ x,
                             const float* __restrict__ k,
                             float* __restrict__ out) {
    __shared__ alignas(16) char smem[2 * TILEB];      // 36,864 B
    const unsigned lds_base = (unsigned)(uintptr_t)(void*)smem; // low 32 bits of flat LDS addr == LDS offset
    const int t = threadIdx.x;
    const long long row0 = (long long)blockIdx.x * TROWS;
    const char* gx = (const char*)(x + row0 * 32);
    const char* gk = (const char*)(k + row0 * 32);
    char*       go = (char*)(out + row0 * 32);

    // Phase 1: async DMA global -> LDS, fully coalesced (16 B/lane contiguous)
#pragma unroll
    for (int i = 0; i < 8; ++i) {
        const int c = i * TROWS + t;                  // chunk 0..1023 (16 B each)
        const int r = c >> 3, w = c & 7;
        const unsigned  loff = (unsigned)(r * PADB + w * 16);
        const long long goff = (long long)c * 16;
        async_copy_g2l_b128(gx + goff, lds_base + loff);
        async_copy_g2l_b128(gk + goff, lds_base + TILEB + loff);
    }
    wait_asynccnt0();
    __syncthreads();

    // Phase 2: each thread owns row t. Pack 32 bit-floats -> 32-bit word.
    // Inputs are exactly 0.0f (0x00000000) or 1.0f (0x3F800000): bit29 == value.
    const uint4* xr = (const uint4*)(smem + t * PADB);
    const uint4* kr = (const uint4*)(smem + TILEB + t * PADB);
    unsigned wx = 0u, wk = 0u;
#pragma unroll
    for (int i = 0; i < 8; ++i) {
        const uint4 a = xr[i];
        const uint4 b = kr[i];
        const int sh = 31 - 4 * i;
        wx |= ((a.x >> 29) & 1u) << sh;
        wx |= ((a.y >> 29) & 1u) << (sh - 1);
        wx |= ((a.z >> 29) & 1u) << (sh - 2);
        wx |= ((a.w >> 29) & 1u) << (sh - 3);
        wk |= ((b.x >> 29) & 1u) << sh;
        wk |= ((b.y >> 29) & 1u) << (sh - 1);
        wk |= ((b.z >> 29) & 1u) << (sh - 2);
        wk |= ((b.w >> 29) & 1u) << (sh - 3);
    }
    const unsigned wout = scale_bits(wx, wk);

    // Unpack result into (now thread-private) x-tile row as raw fp32 words.
    uint4* orow = (uint4*)(smem + t * PADB);
#pragma unroll
    for (int i = 0; i < 8; ++i) {
        const int sh = 31 - 4 * i;
        uint4 f;
        f.x = ((wout >> sh)       & 1u) ? ONE_F32 : 0u;
        f.y = ((wout >> (sh - 1)) & 1u) ? ONE_F32 : 0u;
        f.z = ((wout >> (sh - 2)) & 1u) ? ONE_F32 : 0u;
        f.w = ((wout >> (sh - 3)) & 1u) ? ONE_F32 : 0u;
        orow[i] = f;
    }
    __syncthreads();

    // Phase 3: async DMA LDS -> global, fully coalesced.
#pragma unroll
    for (int i = 0; i < 8; ++i) {
        const int c = i * TROWS + t;
        const int r = c >> 3, w = c & 7;
        const unsigned  loff = (unsigned)(r * PADB + w * 16);
        const long long goff = (long long)c * 16;
        async_copy_l2g_b128(go + goff, lds_base + loff);
    }
    wait_asynccnt0();
}

// Tail path for row counts not divisible by TROWS (unused for N = 2^21).
__global__ void spike_scale_tail_kernel(const float* __restrict__ x,
                                        const float* __restrict__ k,
                                        float* __restrict__ out,
                                        long long start, long long rows) {
    long long r = start + (long long)blockIdx.x * blockDim.x + threadIdx.x;
    if (r >= rows) return;
    const unsigned* xu = (const unsigned*)x;
    const unsigned* ku = (const unsigned*)k;
    unsigned* ou = (unsigned*)out;
    unsigned wx = 0u, wk = 0u;
    for (int j = 0; j < 32; ++j) {
        wx |= ((xu[r * 32 + j] >> 29) & 1u) << (31 - j);
        wk |= ((ku[r * 32 + j] >> 29) & 1u) << (31 - j);
    }
    const unsigned wout = scale_bits(wx, wk);
    for (int j = 0; j < 32; ++j)
        ou[r * 32 + j] = ((wout >> (31 - j)) & 1u) ? ONE_F32 : 0u;
}

extern "C" void kernel_launch(void* const* d_in, const int* in_sizes, int n_in,
                              void* d_out, int out_size, void* d_ws, size_t ws_size,
                              hipStream_t stream) {
    const float* x = (const float*)d_in[0];
    const float* k = (const float*)d_in[1];
    float* out = (float*)d_out;
    const long long rows  = (long long)in_sizes[0] / 32;   // 2,097,152
    const long long nfull = rows / TROWS;
    if (nfull > 0)
        spike_scale_tile_kernel<<<dim3((unsigned)nfull), dim3(TROWS), 0, stream>>>(x, k, out);
    const long long rem = rows - nfull * TROWS;
    if (rem > 0) {
        const long long start = nfull * TROWS;
        spike_scale_tail_kernel<<<dim3((unsigned)((rem + 127) / 128)), dim3(128), 0, stream>>>(
            x, k, out, start, rows);
    }
}